// Encoder_86612310491287
// MI455X (gfx1250) — compile-verified
//
#include <hip/hip_runtime.h>
#include <hip/hip_bf16.h>
#include <math.h>

// Problem constants (match reference)
#define B_DIM   32
#define S_DIM   512
#define D_IN    768
#define D_MID   128
#define D_OUT   4
#define V_DIM   100000
#define N_ROWS  (B_DIM * S_DIM)          // 16384
#define KEY_SPACE (V_DIM * B_DIM)        // 3,200,000 (key = vocab*32 + batch)
#define NUM_SCAN_BLOCKS (KEY_SPACE / 1024) // 3125, each scan block covers 1024 keys

typedef __attribute__((ext_vector_type(2))) float v2f;
typedef __attribute__((ext_vector_type(4))) float v4f;
typedef __attribute__((ext_vector_type(8))) float v8f;

// ---------------------------------------------------------------------------
// K0: zero presence bitmap (u8[KEY_SPACE]) + unique counter
// ---------------------------------------------------------------------------
__global__ void k0_init(unsigned int* presence_u32, unsigned int* U) {
    int i = blockIdx.x * 256 + threadIdx.x;          // grid = 3125*256 = 800000 u32
    presence_u32[i] = 0u;
    if (i == 0) *U = 0u;
}

// ---------------------------------------------------------------------------
// K1: mark present keys. keys.reshape(-1): n = b*S + s, key = vocab*B + b
// ---------------------------------------------------------------------------
__global__ void k1_mark(const int* __restrict__ vocab_ids,
                        unsigned char* __restrict__ presence) {
    int n = blockIdx.x * 256 + threadIdx.x;          // N_ROWS threads
    int b = n >> 9;                                  // S = 512
    int key = vocab_ids[n] * B_DIM + b;
    presence[key] = 1;                               // racy same-value stores OK
}

// ---------------------------------------------------------------------------
// K2a: per-scan-block presence counts (each block covers 1024 presence bytes)
// ---------------------------------------------------------------------------
__global__ void k2a_blocksums(const unsigned char* __restrict__ presence,
                              unsigned int* __restrict__ blockSums) {
    __shared__ unsigned int sm[256];
    int tid = threadIdx.x;
    unsigned int v = ((const unsigned int*)presence)[blockIdx.x * 256 + tid];
    unsigned int cnt = (v * 0x01010101u) >> 24;      // sum of 4 0/1 bytes
    sm[tid] = cnt; __syncthreads();
    for (int off = 128; off > 0; off >>= 1) {
        if (tid < off) sm[tid] += sm[tid + off];
        __syncthreads();
    }
    if (tid == 0) blockSums[blockIdx.x] = sm[0];
}

// ---------------------------------------------------------------------------
// K2b: exclusive scan of the 3125 block sums (single block), total -> U
// ---------------------------------------------------------------------------
__global__ void k2b_scan(const unsigned int* __restrict__ blockSums,
                         unsigned int* __restrict__ scanned,
                         unsigned int* __restrict__ U) {
    __shared__ unsigned int sm[256];
    int tid = threadIdx.x;
    unsigned int running = 0;
    const int iters = (NUM_SCAN_BLOCKS + 255) / 256; // 13
    for (int it = 0; it < iters; ++it) {
        int idx = it * 256 + tid;
        unsigned int val = (idx < NUM_SCAN_BLOCKS) ? blockSums[idx] : 0u;
        sm[tid] = val; __syncthreads();
        for (int off = 1; off < 256; off <<= 1) {    // Hillis-Steele inclusive
            unsigned int t = (tid >= off) ? sm[tid - off] : 0u;
            __syncthreads();
            sm[tid] += t;
            __syncthreads();
        }
        if (idx < NUM_SCAN_BLOCKS) scanned[idx] = running + sm[tid] - val;
        running += sm[255];
        __syncthreads();
    }
    if (tid == 0) *U = running;
}

// ---------------------------------------------------------------------------
// K2c: emit sorted unique keys via local scan + scanned base
// ---------------------------------------------------------------------------
__global__ void k2c_emit(const unsigned char* __restrict__ presence,
                         const unsigned int* __restrict__ scanned,
                         unsigned int* __restrict__ uniq_keys) {
    __shared__ unsigned int sm[256];
    int tid = threadIdx.x;
    unsigned int v = ((const unsigned int*)presence)[blockIdx.x * 256 + tid];
    unsigned int cnt = (v * 0x01010101u) >> 24;
    sm[tid] = cnt; __syncthreads();
    for (int off = 1; off < 256; off <<= 1) {
        unsigned int t = (tid >= off) ? sm[tid - off] : 0u;
        __syncthreads();
        sm[tid] += t;
        __syncthreads();
    }
    unsigned int r = scanned[blockIdx.x] + sm[tid] - cnt;   // exclusive rank
    int keybase = blockIdx.x * 1024 + tid * 4;
    #pragma unroll
    for (int j = 0; j < 4; ++j)
        if ((v >> (8 * j)) & 0xFF) uniq_keys[r++] = (unsigned int)(keybase + j);
}

// ---------------------------------------------------------------------------
// K3: deterministic segment-sum. One 256-thread block per unique slot scans
// the 512 positions of its batch row in ascending s (== segment_sum order).
// Slots >= U get zero rows (matches jnp.unique fill padding).
// ---------------------------------------------------------------------------
__global__ void k3_segsum(const int* __restrict__ vocab_ids,
                          const float* __restrict__ emb,
                          const unsigned int* __restrict__ uniq_keys,
                          const unsigned int* __restrict__ Uptr,
                          float* __restrict__ summed) {
    int r = blockIdx.x;
    int tid = threadIdx.x;
    unsigned int U = *Uptr;
    float a0 = 0.f, a1 = 0.f, a2 = 0.f;
    if (r < (int)U) {
        int k = __builtin_amdgcn_readfirstlane((int)uniq_keys[r]); // scalarize
        int b = k & (B_DIM - 1);
        int v = k >> 5;
        const int*   vrow = vocab_ids + b * S_DIM;
        const float* erow = emb + (size_t)b * S_DIM * D_IN;
        for (int s = 0; s < S_DIM; ++s) {
            if (vrow[s] == v) {                       // uniform branch per block
                const float* e = erow + (size_t)s * D_IN;
                a0 += e[tid]; a1 += e[tid + 256]; a2 += e[tid + 512];
            }
        }
    }
    float* srow = summed + (size_t)r * D_IN;
    srow[tid] = a0; srow[tid + 256] = a1; srow[tid + 512] = a2;
}

// ---------------------------------------------------------------------------
// K4: h = tanh(summed @ W1 + b1) via V_WMMA_F32_16X16X4_F32 (fp32 matrix core).
// 8 waves/block; each wave owns one 16-row strip across all 128 cols
// (8 accumulators reuse one A fragment). B staged through LDS in 64x128
// chunks, stored k-PAIR-INTERLEAVED: {B[k][c], B[k+1][c]} contiguous, so each
// B fragment is one aligned ds_load_b64 into an even VGPR pair (no v_mov
// repacking). Pair-row stride = 288 dwords (288%64==32) -> the two half-waves
// (klo=0 vs klo=2) hit disjoint LDS bank sets.
// A-frag (16x4 f32): lanes 0-15 -> K0,K1 ; lanes 16-31 -> K2,K3 (klo).
// C/D (v8f): VGPR r holds M=r (lanes 0-15) / M=r+8 (lanes 16-31), N=lane&15.
// ---------------------------------------------------------------------------
#define KC 64
#define PAIR_STRIDE 288   // dwords per k-pair row: 128 cols * 2 + 32 pad
__global__ void __launch_bounds__(256) k4_gemm(const float* __restrict__ summed,
                                               const float* __restrict__ W1,
                                               const float* __restrict__ b1,
                                               float* __restrict__ h) {
    __shared__ float Bs[(KC / 2) * PAIR_STRIDE];     // 32*288*4 = 36 KB
    int tid    = threadIdx.x;
    int wave   = tid >> 5;
    int lane   = tid & 31;
    int lane15 = lane & 15;
    int klo    = (lane >> 4) << 1;                   // 0 or 2
    int tileM  = blockIdx.x * 8 + wave;
    const float* arow = summed + (size_t)(tileM * 16 + lane15) * D_IN;

    const v8f zero = {0.f, 0.f, 0.f, 0.f, 0.f, 0.f, 0.f, 0.f};
    v8f acc[8];
    #pragma unroll
    for (int nt = 0; nt < 8; ++nt) acc[nt] = zero;

    for (int kc = 0; kc < D_IN; kc += KC) {
        __syncthreads();
        // cooperative B chunk load + pair-interleave repack:
        // 1024 patches (32 k-pairs x 32 col-groups of 4), 4 per thread
        #pragma unroll
        for (int i = 0; i < 4; ++i) {
            int q = i * 256 + tid;
            int p = q >> 5;                          // k-pair row 0..31
            int c = (q & 31) * 4;                    // col 0..124
            const float* g0 = W1 + (size_t)(kc + 2 * p) * D_MID + c;
            v4f r0 = *(const v4f*)g0;                // row k
            v4f r1 = *(const v4f*)(g0 + D_MID);      // row k+1
            v2f* dst = (v2f*)(Bs + p * PAIR_STRIDE + c * 2);
            v2f t0; t0.x = r0.x; t0.y = r1.x;
            v2f t1; t1.x = r0.y; t1.y = r1.y;
            v2f t2; t2.x = r0.z; t2.y = r1.z;
            v2f t3; t3.x = r0.w; t3.y = r1.w;
            dst[0] = t0; dst[1] = t1; dst[2] = t2; dst[3] = t3;
        }
        if (kc + KC < D_IN)
            __builtin_prefetch(W1 + (size_t)(kc + KC) * D_MID, 0, 1);
        __syncthreads();

        #pragma unroll 4
        for (int k0 = 0; k0 < KC; k0 += 4) {
            v2f a = *(const v2f*)(arow + kc + k0 + klo);       // 8B-aligned b64
            const float* brow = Bs + ((k0 + klo) >> 1) * PAIR_STRIDE;
            #pragma unroll
            for (int nt = 0; nt < 8; ++nt) {
                int col = nt * 16 + lane15;
                v2f bb = *(const v2f*)(brow + col * 2);        // one ds_load_b64
                acc[nt] = __builtin_amdgcn_wmma_f32_16x16x4_f32(
                    false, a, false, bb, (short)0, acc[nt], false, false);
            }
        }
    }

    int rtop = (lane >> 4) * 8;
    #pragma unroll
    for (int nt = 0; nt < 8; ++nt) {
        int col = nt * 16 + lane15;
        float bias = b1[col];
        #pragma unroll
        for (int rr = 0; rr < 8; ++rr) {
            int row = tileM * 16 + rr + rtop;
            h[(size_t)row * D_MID + col] = tanhf(acc[nt][rr] + bias);
        }
    }
}

// ---------------------------------------------------------------------------
// K5: encoded = sigmoid(h[r] . encoder_weights[vocab[r]]); write ids + encoded.
// One 128-thread block per row; gathered b128 weight loads; LDS tree reduce.
// ---------------------------------------------------------------------------
__global__ void k5_encode(const float* __restrict__ h,
                          const float* __restrict__ encw,
                          const unsigned int* __restrict__ uniq_keys,
                          const unsigned int* __restrict__ Uptr,
                          float* __restrict__ out) {
    __shared__ float red[D_MID * 4];
    int r = blockIdx.x;
    int tid = threadIdx.x;
    unsigned int U = *Uptr;
    int k = (r < (int)U) ? (int)uniq_keys[r] : 0;    // fill slots -> key 0
    k = __builtin_amdgcn_readfirstlane(k);
    int b = k & (B_DIM - 1);
    int v = k >> 5;

    float hi = h[(size_t)r * D_MID + tid];
    v4f wv = *(const v4f*)(encw + (size_t)v * (D_MID * D_OUT) + tid * 4);
    red[tid * 4 + 0] = hi * wv.x;
    red[tid * 4 + 1] = hi * wv.y;
    red[tid * 4 + 2] = hi * wv.z;
    red[tid * 4 + 3] = hi * wv.w;
    __syncthreads();
    for (int off = 64; off > 0; off >>= 1) {
        if (tid < off) {
            red[tid * 4 + 0] += red[(tid + off) * 4 + 0];
            red[tid * 4 + 1] += red[(tid + off) * 4 + 1];
            red[tid * 4 + 2] += red[(tid + off) * 4 + 2];
            red[tid * 4 + 3] += red[(tid + off) * 4 + 3];
        }
        __syncthreads();
    }
    if (tid < 4) {
        float dot = red[tid];
        out[(size_t)N_ROWS * 2 + (size_t)r * 4 + tid] = 1.0f / (1.0f + expf(-dot));
    }
    if (tid == 0) {                                  // ids (N,2), value-cast to f32
        out[(size_t)r * 2 + 0] = (float)v;
        out[(size_t)r * 2 + 1] = (float)b;
    }
}

// ---------------------------------------------------------------------------
extern "C" void kernel_launch(void* const* d_in, const int* in_sizes, int n_in,
                              void* d_out, int out_size, void* d_ws, size_t ws_size,
                              hipStream_t stream) {
    const int*   vocab_ids  = (const int*)d_in[0];
    const float* embeddings = (const float*)d_in[1];
    const float* W1         = (const float*)d_in[2];
    const float* b1         = (const float*)d_in[3];
    const float* encw       = (const float*)d_in[4];
    float*       out        = (float*)d_out;
    char*        ws         = (char*)d_ws;

    // workspace carve-out (~62 MB total), 256B aligned
    size_t off = 0;
    auto carve = [&](size_t bytes) {
        size_t p = off;
        off = (off + bytes + 255) & ~(size_t)255;
        return p;
    };
    unsigned char* presence  = (unsigned char*)(ws + carve(KEY_SPACE));
    unsigned int*  blockSums = (unsigned int*)(ws + carve(NUM_SCAN_BLOCKS * 4));
    unsigned int*  scanned   = (unsigned int*)(ws + carve(NUM_SCAN_BLOCKS * 4));
    unsigned int*  Uc        = (unsigned int*)(ws + carve(4));
    unsigned int*  uniq      = (unsigned int*)(ws + carve(N_ROWS * 4));
    float*         summed    = (float*)(ws + carve((size_t)N_ROWS * D_IN * 4));
    float*         hbuf      = (float*)(ws + carve((size_t)N_ROWS * D_MID * 4));

    k0_init      <<<NUM_SCAN_BLOCKS, 256, 0, stream>>>((unsigned int*)presence, Uc);
    k1_mark      <<<N_ROWS / 256,    256, 0, stream>>>(vocab_ids, presence);
    k2a_blocksums<<<NUM_SCAN_BLOCKS, 256, 0, stream>>>(presence, blockSums);
    k2b_scan     <<<1,               256, 0, stream>>>(blockSums, scanned, Uc);
    k2c_emit     <<<NUM_SCAN_BLOCKS, 256, 0, stream>>>(presence, scanned, uniq);
    k3_segsum    <<<N_ROWS,          256, 0, stream>>>(vocab_ids, embeddings, uniq, Uc, summed);
    k4_gemm      <<<N_ROWS / 16 / 8, 256, 0, stream>>>(summed, W1, b1, hbuf);
    k5_encode    <<<N_ROWS,          128, 0, stream>>>(hbuf, encw, uniq, Uc, out);
}